// VectorMSE_71949292142895
// MI455X (gfx1250) — compile-verified
//
#include <hip/hip_runtime.h>
#include <hip/hip_bf16.h>
#include <math.h>

typedef __attribute__((ext_vector_type(2))) float v2f;
typedef __attribute__((ext_vector_type(8))) float v8f;

#define BDIM 512
#define NDIM 128
#define TILES_PER_ROW (BDIM / 16)   // 32
#define WAVES_PER_BLOCK 4

// One wave (32 lanes) computes one 16x16 tile of norms[i,j] = mean_n ||gt[j,n]-pred[i,n]||.
// Per n: V_WMMA_F32_16X16X4_F32 computes  D[i,j] = (-2 p_i) . g_j + ||p_i||^2   (K=3 padded
// with the ||p||^2 * 1 term in the K=4 slot), then each lane adds ||g_j||^2 (its own column's
// value), clamps, takes raw-hw v_sqrt_f32, and accumulates into the 8-elem f32 fragment.
__global__ void __launch_bounds__(32 * WAVES_PER_BLOCK)
pairwise_norm_kernel(const float* __restrict__ pred,
                     const float* __restrict__ gt,
                     float* __restrict__ norms) {
  const int lane = threadIdx.x & 31;     // 0..31, wave32
  const int wave = threadIdx.x >> 5;     // 0..3
  const int tile = blockIdx.x * WAVES_PER_BLOCK + wave;   // 0..1023
  const int ti   = tile / TILES_PER_ROW; // tile row  (i / 16)
  const int tj   = tile % TILES_PER_ROW; // tile col  (j / 16)

  const int m    = lane & 15;            // row/col within tile held by this lane
  const bool hi  = lane >= 16;           // lane group selects K pair (A/B), M offset (C/D)

  const int rowA = ti * 16 + m;          // pred row this lane feeds into A
  const int colB = tj * 16 + m;          // gt   row this lane feeds into B

  const float* __restrict__ pbase = pred + (size_t)rowA * (NDIM * 3);
  const float* __restrict__ gbase = gt   + (size_t)colB * (NDIM * 3);

  v8f acc = {0.f, 0.f, 0.f, 0.f, 0.f, 0.f, 0.f, 0.f};

#pragma unroll 4
  for (int n = 0; n < NDIM; ++n) {
    const float* pp = pbase + n * 3;
    const float* gp = gbase + n * 3;
    const float px = pp[0], py = pp[1], pz = pp[2];
    const float gx = gp[0], gy = gp[1], gz = gp[2];

    const float pn = __builtin_fmaf(px, px, __builtin_fmaf(py, py, pz * pz));
    const float gn = __builtin_fmaf(gx, gx, __builtin_fmaf(gy, gy, gz * gz));

    // A-matrix 16x4 f32 layout: lanes 0-15 hold (K0,K1), lanes 16-31 hold (K2,K3), M = lane%16.
    v2f a;
    a.x = hi ? (-2.0f * pz) : (-2.0f * px);
    a.y = hi ? pn           : (-2.0f * py);
    // B-matrix 4x16 f32 layout (mirrored): lanes 0-15 rows (K0,K1), lanes 16-31 rows (K2,K3).
    v2f b;
    b.x = hi ? gz   : gx;
    b.y = hi ? 1.0f : gy;

    v8f c = {0.f, 0.f, 0.f, 0.f, 0.f, 0.f, 0.f, 0.f};
    // D = A x B + 0 :  D[i,j] = -2 p_i . g_j + ||p_i||^2
    v8f d = __builtin_amdgcn_wmma_f32_16x16x4_f32(
        /*neg_a=*/false, a, /*neg_b=*/false, b,
        /*c_mod=*/(short)0, c, /*reuse_a=*/false, /*reuse_b=*/false);

#pragma unroll
    for (int r = 0; r < 8; ++r) {
      float t = d[r] + gn;               // ||p_i - g_j||^2 ; gn is this lane's column term
      t = fmaxf(t, 0.0f);                // guard tiny negative from cancellation
      acc[r] += __builtin_amdgcn_sqrtf(t);  // raw v_sqrt_f32 (~1 ulp), no IEEE fixup chain
    }
  }

  // C/D fragment layout: VGPR r holds M = r + 8*(lane>=16), N = lane%16.
  const float invN = 1.0f / (float)NDIM;
  const int base_m = hi ? 8 : 0;
  const int col    = tj * 16 + m;
#pragma unroll
  for (int r = 0; r < 8; ++r) {
    const int row = ti * 16 + base_m + r;
    norms[(size_t)row * BDIM + col] = acc[r] * invN;
  }
}

// One block per row i: logits[i,j] = -0.5 * norms[i,j]^2 / sigma^2,
// loss[i] = (logsumexp_j - logits[i,i]) * 2*sigma^2 * w[i].
__global__ void __launch_bounds__(256)
loss_kernel(const float* __restrict__ norms,
            const float* __restrict__ weights,
            const float* __restrict__ sigma,
            float* __restrict__ out) {
  const int i = blockIdx.x;
  const int t = threadIdx.x;             // 0..255, each handles j = t and t+256

  const float sig       = sigma[0];
  const float noise_var = sig * sig;
  const float scale     = -0.5f / noise_var;

  const float* __restrict__ row = norms + (size_t)i * BDIM;

  float n0 = row[t];
  float n1 = row[t + 256];
  const float l0 = scale * n0 * n0;
  const float l1 = scale * n1 * n1;

  __shared__ float smax[256];
  __shared__ float ssum[256];

  smax[t] = fmaxf(l0, l1);
  __syncthreads();
#pragma unroll
  for (int s = 128; s > 0; s >>= 1) {
    if (t < s) smax[t] = fmaxf(smax[t], smax[t + s]);
    __syncthreads();
  }
  const float gmax = smax[0];
  __syncthreads();

  ssum[t] = __expf(l0 - gmax) + __expf(l1 - gmax);
  __syncthreads();
#pragma unroll
  for (int s = 128; s > 0; s >>= 1) {
    if (t < s) ssum[t] += ssum[t + s];
    __syncthreads();
  }

  if (t == 0) {
    const float nd    = row[i];
    const float ldiag = scale * nd * nd;
    const float lse   = gmax + __logf(ssum[0]);
    out[i] = (lse - ldiag) * (2.0f * noise_var) * weights[i];
  }
}

extern "C" void kernel_launch(void* const* d_in, const int* in_sizes, int n_in,
                              void* d_out, int out_size, void* d_ws, size_t ws_size,
                              hipStream_t stream) {
  const float* pred    = (const float*)d_in[0];  // [512,128,3] f32
  const float* gt      = (const float*)d_in[1];  // [512,128,3] f32
  const float* weights = (const float*)d_in[2];  // [512]       f32
  const float* sigma   = (const float*)d_in[3];  // scalar      f32
  float* out   = (float*)d_out;                  // [512]       f32
  float* norms = (float*)d_ws;                   // [512,512]   f32 scratch (1 MB)

  const int n_tiles = TILES_PER_ROW * TILES_PER_ROW;            // 1024
  pairwise_norm_kernel<<<n_tiles / WAVES_PER_BLOCK, 32 * WAVES_PER_BLOCK, 0, stream>>>(
      pred, gt, norms);
  loss_kernel<<<BDIM, 256, 0, stream>>>(norms, weights, sigma, out);
}